// GATv2Layer_42932493091197
// MI455X (gfx1250) — compile-verified
//
#include <hip/hip_runtime.h>
#include <hip/hip_bf16.h>

#define N_NODES   50000
#define N_EDGES   800000
#define E_TOT     850000      // N_EDGES + N_NODES self loops
#define IN_F      128
#define HC        128         // HEADS * OUT_F
#define NEG_SLOPE 0.2f

typedef __attribute__((ext_vector_type(2))) float v2f;
typedef __attribute__((ext_vector_type(8))) float v8f;

// ---------------------------------------------------------------------------
// Init: out[n][hc] = bias[hc];  m = -FLT_MAX;  denom = 0
// ---------------------------------------------------------------------------
__global__ void gat_init(float* __restrict__ out, const float* __restrict__ bias,
                         float* __restrict__ m, float* __restrict__ denom) {
  int i = blockIdx.x * blockDim.x + threadIdx.x;
  if (i < N_NODES * HC) out[i] = bias[i & (HC - 1)];
  if (i < N_NODES * 4) {
    m[i] = -3.402823466e38f;
    denom[i] = 0.0f;
  }
}

// ---------------------------------------------------------------------------
// GEMM: xl = x @ W_l, xr = x @ W_r via V_WMMA_F32_16X16X4_F32.
// One wave per 16x16 output tile. 3125 M-tiles x 16 N-tiles (8 for W_l,
// 8 for W_r) = 50000 waves, launched exactly (8 waves / 256-thread block).
// K=128 -> 32 WMMA steps per tile.
//
// A 16x4 f32 fragment (2 VGPRs): lane&15 = M row; lane>>4 selects K pair
//   {0,1} vs {2,3}.  B 4x16 fragment mirrors it (lane&15 = N col).
// C/D 16x16 f32: VGPR v, row = v + 8*(lane>>4), col = lane&15.
// ---------------------------------------------------------------------------
__global__ void __launch_bounds__(256)
gat_gemm(const float* __restrict__ x, const float* __restrict__ Wl,
         const float* __restrict__ Wr, float* __restrict__ xl,
         float* __restrict__ xr) {
  const int gw   = (blockIdx.x * blockDim.x + threadIdx.x) >> 5; // global wave
  const int lane = threadIdx.x & 31;
  const int mt = gw >> 4;        // 0..3124 (exact: 3125*16 = 50000 rows)
  const int nt = gw & 15;        // 0..15

  const float* __restrict__ W   = (nt < 8) ? Wl : Wr;
  float* __restrict__       dst = (nt < 8) ? xl : xr;
  const int ncol  = (nt & 7) * 16 + (lane & 15);   // column in [0,128)
  const int mrow  = mt * 16 + (lane & 15);
  const int khalf = (lane >> 4) * 2;               // 0 or 2

  const float* __restrict__ xrow = x + (size_t)mrow * IN_F;

  v8f c = {};
#pragma unroll
  for (int kb = 0; kb < IN_F; kb += 4) {
    v2f a, b;
    // A: x[mrow][kb+khalf .. +1]  (8B-aligned pair)
    const float2 ax = *(const float2*)(xrow + kb + khalf);
    a.x = ax.x; a.y = ax.y;
    // B: W[kb+khalf][ncol], W[kb+khalf+1][ncol]
    b.x = W[(size_t)(kb + khalf) * HC + ncol];
    b.y = W[(size_t)(kb + khalf + 1) * HC + ncol];
    c = __builtin_amdgcn_wmma_f32_16x16x4_f32(false, a, false, b,
                                              (short)0, c, false, false);
  }

  const int mbase = mt * 16 + 8 * (lane >> 4);
#pragma unroll
  for (int v = 0; v < 8; ++v)
    dst[(size_t)(mbase + v) * HC + ncol] = c[v];
}

// ---------------------------------------------------------------------------
// Pass A: per-edge GATv2 logits + segment-max over dst.
// One wave per edge. Lane l covers channels 4l..4l+3 (float4 load), which all
// belong to head h = l>>3; 8-lane xor-shuffle reduction yields per-head logit.
// ---------------------------------------------------------------------------
__global__ void __launch_bounds__(256)
gat_logits(const int* __restrict__ ei, const float* __restrict__ xl,
           const float* __restrict__ xr, const float* __restrict__ att,
           float* __restrict__ logits, float* __restrict__ m) {
  const unsigned gid  = blockIdx.x * blockDim.x + threadIdx.x;
  const unsigned e    = gid >> 5;            // exact: 850000 waves launched
  const unsigned lane = threadIdx.x & 31;

  int src, dst;
  if (e < N_EDGES) { src = ei[e]; dst = ei[N_EDGES + e]; }
  else             { src = dst = (int)(e - N_EDGES); }

  const float4 a = ((const float4*)(xl + (size_t)src * HC))[lane];
  const float4 b = ((const float4*)(xr + (size_t)dst * HC))[lane];
  const float4 w = ((const float4*)att)[lane];

  float vx = a.x + b.x, vy = a.y + b.y, vz = a.z + b.z, vw = a.w + b.w;
  vx = (vx > 0.f) ? vx : vx * NEG_SLOPE;
  vy = (vy > 0.f) ? vy : vy * NEG_SLOPE;
  vz = (vz > 0.f) ? vz : vz * NEG_SLOPE;
  vw = (vw > 0.f) ? vw : vw * NEG_SLOPE;
  float s = vx * w.x + vy * w.y + vz * w.z + vw * w.w;

  // reduce over the 8 lanes sharing one head
  s += __shfl_xor(s, 1);
  s += __shfl_xor(s, 2);
  s += __shfl_xor(s, 4);

  if ((lane & 7) == 0) {
    const int h = lane >> 3;
    logits[(size_t)e * 4 + h] = s;
    atomicMax(&m[(size_t)dst * 4 + h], s);   // global_atomic_max_num_f32
  }
}

// ---------------------------------------------------------------------------
// Pass B: denom[dst][h] += exp(logit - m[dst][h]).  One thread per (edge,head).
// ---------------------------------------------------------------------------
__global__ void gat_expsum(const int* __restrict__ ei,
                           const float* __restrict__ logits,
                           const float* __restrict__ m,
                           float* __restrict__ denom) {
  const unsigned gid = blockIdx.x * blockDim.x + threadIdx.x;
  if (gid >= (unsigned)E_TOT * 4) return;
  const unsigned e = gid >> 2;
  const unsigned h = gid & 3;

  int dst;
  if (e < N_EDGES) dst = ei[N_EDGES + e];
  else             dst = (int)(e - N_EDGES);

  const float ex = __expf(logits[(size_t)e * 4 + h] - m[(size_t)dst * 4 + h]);
  atomicAdd(&denom[(size_t)dst * 4 + h], ex);
}

// ---------------------------------------------------------------------------
// Pass C: out[dst] += alpha * xl[src].  One wave per edge, float4 per lane,
// alpha recomputed from stored logit + m + denom (all small / L2-resident).
// ---------------------------------------------------------------------------
__global__ void __launch_bounds__(256)
gat_aggregate(const int* __restrict__ ei, const float* __restrict__ xl,
              const float* __restrict__ logits, const float* __restrict__ m,
              const float* __restrict__ denom, float* __restrict__ out) {
  const unsigned gid  = blockIdx.x * blockDim.x + threadIdx.x;
  const unsigned e    = gid >> 5;            // exact: 850000 waves launched
  const unsigned lane = threadIdx.x & 31;

  int src, dst;
  if (e < N_EDGES) { src = ei[e]; dst = ei[N_EDGES + e]; }
  else             { src = dst = (int)(e - N_EDGES); }

  const int h = lane >> 3;
  const float alpha =
      __expf(logits[(size_t)e * 4 + h] - m[(size_t)dst * 4 + h]) /
      (denom[(size_t)dst * 4 + h] + 1e-16f);

  const float4 v = ((const float4*)(xl + (size_t)src * HC))[lane];
  float* po = out + (size_t)dst * HC + lane * 4;
  atomicAdd(po + 0, v.x * alpha);
  atomicAdd(po + 1, v.y * alpha);
  atomicAdd(po + 2, v.z * alpha);
  atomicAdd(po + 3, v.w * alpha);
}

// ---------------------------------------------------------------------------
extern "C" void kernel_launch(void* const* d_in, const int* in_sizes, int n_in,
                              void* d_out, int out_size, void* d_ws, size_t ws_size,
                              hipStream_t stream) {
  const float* x    = (const float*)d_in[0];
  const int*   ei   = (const int*)  d_in[1];   // [2, N_EDGES] flat
  const float* Wl   = (const float*)d_in[2];
  const float* Wr   = (const float*)d_in[3];
  const float* att  = (const float*)d_in[4];
  const float* bias = (const float*)d_in[5];
  float* out = (float*)d_out;

  // workspace layout (floats)
  float* ws     = (float*)d_ws;
  float* xl     = ws;                                   // 50000*128
  float* xr     = xl + (size_t)N_NODES * HC;            // 50000*128
  float* m      = xr + (size_t)N_NODES * HC;            // 50000*4
  float* denom  = m + (size_t)N_NODES * 4;              // 50000*4
  float* logits = denom + (size_t)N_NODES * 4;          // 850000*4

  // init out=bias, m=-inf, denom=0
  gat_init<<<(N_NODES * HC + 255) / 256, 256, 0, stream>>>(out, bias, m, denom);

  // xl/xr GEMMs: 3125*16 = 50000 waves = 6250 blocks of 8 waves
  gat_gemm<<<6250, 256, 0, stream>>>(x, Wl, Wr, xl, xr);

  // edge passes: 850000 waves = 106250 blocks (exact)
  gat_logits<<<106250, 256, 0, stream>>>(ei, xl, xr, att, logits, m);
  gat_expsum<<<((unsigned)E_TOT * 4 + 255) / 256, 256, 0, stream>>>(ei, logits, m, denom);
  gat_aggregate<<<106250, 256, 0, stream>>>(ei, xl, logits, m, denom, out);
}